// MicrofacetBase_19439021981995
// MI455X (gfx1250) — compile-verified
//
#include <hip/hip_runtime.h>
#include <math.h>

// ---------------- tiling constants ----------------
#define THREADS            256
#define PPT                4                         // points per thread
#define TILE_PTS           (THREADS * PPT)           // 1024 points / block
#define FLOATS_PER_PT      6                         // light(3) + view(3)
#define TILE_FLOATS        (TILE_PTS * FLOATS_PER_PT)// 6144 floats = 24576 B
#define TILE_BYTES         (TILE_FLOATS * 4)         // 24576
#define CHUNKS16           (TILE_BYTES / 16)         // 1536 x 16B chunks
#define CHUNKS_PER_THREAD  (CHUNKS16 / THREADS)      // 6

typedef int v4i_gcc __attribute__((vector_size(4 * sizeof(int))));

// ---- gfx1250 async global->LDS copy (ASYNCcnt path) ----
__device__ __forceinline__ void async_ld16(const void* g, const void* l) {
#if defined(__HIP_DEVICE_COMPILE__)
  auto gp = (__attribute__((address_space(1))) v4i_gcc*)(unsigned long long)g;
  auto lp = (__attribute__((address_space(3))) v4i_gcc*)(unsigned)(unsigned long long)l;
#if __has_builtin(__builtin_amdgcn_global_load_async_to_lds_b128)
  __builtin_amdgcn_global_load_async_to_lds_b128(gp, lp, /*offset=*/0, /*cpol=*/0);
#else
  unsigned loff = (unsigned)(unsigned long long)lp;
  asm volatile("global_load_async_to_lds_b128 %0, %1, off"
               :: "v"(loff), "v"(gp) : "memory");
#endif
#else
  (void)g; (void)l;
#endif
}

__device__ __forceinline__ void wait_async0() {
#if defined(__HIP_DEVICE_COMPILE__)
#if __has_builtin(__builtin_amdgcn_s_wait_asynccnt)
  __builtin_amdgcn_s_wait_asynccnt(0);
#else
  asm volatile("s_wait_asynccnt 0" ::: "memory");
#endif
#endif
}

// ---- fast hardware math (<=2 ulp; kernel is HBM-bound, keep VALU light) ----
__device__ __forceinline__ float frcp(float x)  { return __builtin_amdgcn_rcpf(x); }
__device__ __forceinline__ float frsq(float x)  { return __builtin_amdgcn_rsqf(x); }
__device__ __forceinline__ float fsqrt(float x) { return __builtin_amdgcn_sqrtf(x); }
// base_color > 0, so pow(x,2.2) == exp2(2.2*log2(x)); v_log_f32 + v_exp_f32.
__device__ __forceinline__ float pow22(float x) {
#if __has_builtin(__builtin_amdgcn_exp2f) && __has_builtin(__builtin_amdgcn_logf)
  return __builtin_amdgcn_exp2f(2.2f * __builtin_amdgcn_logf(x));
#else
  return exp2f(2.2f * log2f(x));   // host pass / older toolchain fallback
#endif
}

// Per-point BRDF. Uniform precomputes: a2opi = alpha^2/pi, a2m1 = alpha^2-1,
// eta2m1 = eta^2-1. Note G = cos_nl*cos_nv cancels exactly against the
// 4*cos_nl*cos_nv denominator -> scale = 0.25*D*F.
__device__ __forceinline__ void brdf_point(const float* q, float a2opi, float a2m1,
                                           float eta2m1,
                                           float lin0, float lin1, float lin2,
                                           float* r) {
  float lx = q[0], ly = q[1], lz = q[2];
  float vx = q[3], vy = q[4], vz = q[5];
  float hx = lx + vx, hy = ly + vy, hz = lz + vz;
  float inv = frsq(hx*hx + hy*hy + hz*hz);          // v_rsq_f32
  float cos_nh = hz * inv;
  float c = (hx*vx + hy*vy + hz*vz) * inv;          // cos_hv
  float dd = fmaf(cos_nh * cos_nh, a2m1, 1.0f);
  float D = a2opi * frcp(dd * dd);                  // a2/(pi*dd*dd)
  float g2 = fmaf(c, c, eta2m1);                    // eta^2 + c^2 - 1
  float gq = fsqrt(fmaxf(g2, 1e-12f));
  float gpc = gq + c, gmc = gq - c;
  float aa = gmc * frcp(gpc);
  float bb = fmaf(c, gpc, -1.0f) * frcp(fmaf(c, gmc, 1.0f));
  float F = (g2 > 0.0f) ? 0.5f * aa * aa * fmaf(bb, bb, 1.0f) : 1.0f;
  float scale = 0.25f * D * F;                      // (D*G*F)/(4*cos_nl*cos_nv)
  r[0] = lin0 * scale; r[1] = lin1 * scale; r[2] = lin2 * scale;
}

__global__ __launch_bounds__(THREADS)
void microfacet_stream_kernel(const float* __restrict__ in,
                              const float* __restrict__ base_color,
                              const float* __restrict__ alpha,
                              const float* __restrict__ eta,
                              float* __restrict__ out,
                              long long nPts) {
  __shared__ __align__(16) float tile[TILE_FLOATS];
  const int tid = threadIdx.x;
  const long long tileStart = (long long)blockIdx.x * TILE_PTS;

  if (tileStart + TILE_PTS <= nPts) {
    // ---- stage 24KB input tile to LDS via async DMA (b128 chunks) ----
    const char* gbase = (const char*)in + tileStart * (FLOATS_PER_PT * 4);
    char* lbase = (char*)tile;
#pragma unroll
    for (int i = 0; i < CHUNKS_PER_THREAD; ++i) {
      int c = (i * THREADS + tid) * 16;
      async_ld16(gbase + c, lbase + c);
    }

    // ---- overlap uniform params + base_color^2.2 with the DMA ----
    float a1 = alpha[0];
    float a2 = a1 * a1;
    float a2m1 = a2 - 1.0f;
    float a2opi = a2 * 0.31830988618379067154f;     // a2 * (1/pi)
    float e0 = eta[0];
    float eta2m1 = fmaf(e0, e0, -1.0f);
    float lin0 = pow22(base_color[0]);
    float lin1 = pow22(base_color[1]);
    float lin2 = pow22(base_color[2]);

    wait_async0();      // drain this wave's ASYNCcnt
    __syncthreads();    // publish LDS tile across all 8 waves

    // ---- 96B contiguous LDS read per thread (6 x ds_load_b128) ----
    float q[PPT * FLOATS_PER_PT];
    const float4* ls = (const float4*)(tile + tid * (PPT * FLOATS_PER_PT));
#pragma unroll
    for (int i = 0; i < (PPT * FLOATS_PER_PT) / 4; ++i) {
      float4 v = ls[i];
      q[4*i+0] = v.x; q[4*i+1] = v.y; q[4*i+2] = v.z; q[4*i+3] = v.w;
    }

    float r[PPT * 3];
#pragma unroll
    for (int j = 0; j < PPT; ++j)
      brdf_point(q + j * FLOATS_PER_PT, a2opi, a2m1, eta2m1,
                 lin0, lin1, lin2, r + j * 3);

    // ---- 48B contiguous store per thread (3 x global_store_b128) ----
    float4* dst = (float4*)(out + (tileStart + (long long)tid * PPT) * 3);
#pragma unroll
    for (int i = 0; i < (PPT * 3) / 4; ++i)
      dst[i] = make_float4(r[4*i+0], r[4*i+1], r[4*i+2], r[4*i+3]);
  } else {
    // ---- tail tile (unused for N = 4M, kept for robustness) ----
    float a1 = alpha[0];
    float a2 = a1 * a1;
    float a2m1 = a2 - 1.0f;
    float a2opi = a2 * 0.31830988618379067154f;
    float e0 = eta[0];
    float eta2m1 = fmaf(e0, e0, -1.0f);
    float lin0 = pow22(base_color[0]);
    float lin1 = pow22(base_color[1]);
    float lin2 = pow22(base_color[2]);
    for (int j = 0; j < PPT; ++j) {
      long long p = tileStart + tid + (long long)j * THREADS;
      if (p < nPts) {
        float q[6];
#pragma unroll
        for (int k = 0; k < 6; ++k) q[k] = in[p * 6 + k];
        float r[3];
        brdf_point(q, a2opi, a2m1, eta2m1, lin0, lin1, lin2, r);
        out[p*3+0] = r[0]; out[p*3+1] = r[1]; out[p*3+2] = r[2];
      }
    }
  }
}

extern "C" void kernel_launch(void* const* d_in, const int* in_sizes, int n_in,
                              void* d_out, int out_size, void* d_ws, size_t ws_size,
                              hipStream_t stream) {
  const float* in         = (const float*)d_in[0];   // (N,2,3) f32
  const float* base_color = (const float*)d_in[1];   // (3,)
  const float* alpha      = (const float*)d_in[2];   // (1,)
  const float* eta        = (const float*)d_in[3];   // (1,)
  float* out = (float*)d_out;                        // (N,3) f32

  long long nPts = (long long)in_sizes[0] / 6;
  int blocks = (int)((nPts + TILE_PTS - 1) / TILE_PTS);   // 4096 for N=4M
  microfacet_stream_kernel<<<blocks, THREADS, 0, stream>>>(
      in, base_color, alpha, eta, out, nPts);
}